// GCN_23149873725488
// MI455X (gfx1250) — compile-verified
//
#include <hip/hip_runtime.h>

#define N_NODES   100000
#define N_EDGES   1600000
#define F_IN      128
#define HIDDEN    64
#define N_CLASSES 32

typedef float v2f __attribute__((ext_vector_type(2)));
typedef float v8f __attribute__((ext_vector_type(8)));

// ---------------------------------------------------------------- utilities
__global__ void fill_kernel(float* __restrict__ p, float v, int n) {
    int i = blockIdx.x * blockDim.x + threadIdx.x;
    if (i < n) p[i] = v;
}

__global__ void degree_kernel(const int* __restrict__ dst, float* __restrict__ deg, int E) {
    int e = blockIdx.x * blockDim.x + threadIdx.x;
    if (e < E) atomicAdd(&deg[dst[e]], 1.0f);
}

__global__ void rsqrt_kernel(float* __restrict__ d, int n) {
    int i = blockIdx.x * blockDim.x + threadIdx.x;
    if (i < n) {
        float v = d[i];
        d[i] = (v > 0.0f) ? rsqrtf(v) : 0.0f;   // deg >= 1 always (self-loop), guard matches ref
    }
}

// ---------------------------------------------------------------- WMMA GEMM
// C[M,N] = A[M,K] * W[K,N], row-major, f32 via V_WMMA_F32_16X16X4_F32.
// Block = 256 threads = 8 waves; each wave owns one 16x16 output tile.
// A tile staged in LDS with +4 float row padding (A-fragment reads stride K
// floats -> padded stride K+4 gives conflict-free banks). W fully in LDS.
template <int K, int N>
__global__ __launch_bounds__(256)
void gemm_wmma_kernel(const float* __restrict__ A, const float* __restrict__ W,
                      float* __restrict__ C, int M) {
    constexpr int NT    = N / 16;     // n-tiles covered by the 8 waves
    constexpr int MT    = 8 / NT;     // m-tiles per block
    constexpr int ROWS  = MT * 16;    // rows of A per block
    constexpr int APITCH = K + 4;     // padded LDS row pitch (floats)

    __shared__ float sA[ROWS * APITCH];
    __shared__ float sW[K * N];

    const int tid     = threadIdx.x;
    const int rowBase = blockIdx.x * ROWS;

    // Stage A tile (ROWS x K), float4 per transaction, zero-fill OOB rows.
    for (int i = tid; i < (ROWS * K) / 4; i += 256) {
        int e  = i * 4;
        int r  = e / K;
        int c  = e % K;
        int gr = rowBase + r;
        float4 v = make_float4(0.f, 0.f, 0.f, 0.f);
        if (gr < M) v = *reinterpret_cast<const float4*>(A + (size_t)gr * K + c);
        *reinterpret_cast<float4*>(sA + r * APITCH + c) = v;
    }
    // Stage W (K x N).
    for (int i = tid; i < (K * N) / 4; i += 256) {
        *reinterpret_cast<float4*>(sW + i * 4) =
            *reinterpret_cast<const float4*>(W + i * 4);
    }
    __syncthreads();

    const int wave = tid >> 5;
    const int lane = tid & 31;
    const int mt   = wave / NT;
    const int nt   = wave % NT;
    const int half = lane >> 4;   // 0: lanes 0-15, 1: lanes 16-31
    const int l15  = lane & 15;

    // ISA 7.12.2 layouts for 16x16x4 f32:
    //   A frag: lane (half,l15), VGPR v holds A[M=l15][K = kbase + 2*half + v]
    //   B frag: VGPR v holds W[K = kbase + 2*half + v][N = l15]
    //   C/D:    VGPR v holds C[M = v + 8*half][N = l15]
    const float* aRow = sA + (mt * 16 + l15) * APITCH;
    const float* wCol = sW + nt * 16 + l15;

    v8f acc = {};
    for (int k = 0; k < K; k += 4) {
        int kk = k + 2 * half;
        v2f a;
        a.x = aRow[kk];
        a.y = aRow[kk + 1];
        v2f b;
        b.x = wCol[kk * N];
        b.y = wCol[(kk + 1) * N];
        acc = __builtin_amdgcn_wmma_f32_16x16x4_f32(
            /*neg_a=*/false, a, /*neg_b=*/false, b,
            /*c_mod=*/(short)0, acc, /*reuse_a=*/false, /*reuse_b=*/false);
    }

#pragma unroll
    for (int v = 0; v < 8; ++v) {
        int m  = mt * 16 + v + 8 * half;
        int gr = rowBase + m;
        if (gr < M) C[(size_t)gr * N + nt * 16 + l15] = acc[v];
    }
}

// ------------------------------------------------------- edge scatter (agg)
// One wave per edge; 32 lanes stripe the feature dim. F is 64 or 32.
template <int F>
__global__ __launch_bounds__(256)
void scatter_kernel(const int* __restrict__ ei, const float* __restrict__ h,
                    const float* __restrict__ dinv, float* __restrict__ agg) {
    int t    = blockIdx.x * blockDim.x + threadIdx.x;
    int e    = t >> 5;             // wave-uniform
    if (e >= N_EDGES) return;
    int lane = t & 31;
    int s = ei[e];
    int d = ei[N_EDGES + e];
    float norm = dinv[s] * dinv[d];
    const float* hs = h + (size_t)s * F;
    float*       ad = agg + (size_t)d * F;
    atomicAdd(&ad[lane], hs[lane] * norm);
    if (F == 64) atomicAdd(&ad[lane + 32], hs[lane + 32] * norm);
}

// ---------------------------------------------- layer-1 finalize: +self,+b,ReLU
__global__ void finalize1_kernel(const float* __restrict__ h, const float* __restrict__ dinv,
                                 const float* __restrict__ b1, float* __restrict__ agg) {
    int t = blockIdx.x * blockDim.x + threadIdx.x;
    if (t >= N_NODES * HIDDEN) return;
    int i = t >> 6;        // / 64
    int f = t & 63;
    float di = dinv[i];
    float v  = agg[t] + h[t] * di * di + b1[f];
    agg[t]   = v > 0.0f ? v : 0.0f;   // in-place -> becomes layer-2 input
}

// --------------------------- layer-2 finalize + log_softmax (1 wave per node)
__global__ __launch_bounds__(256)
void finalize2_logsoftmax_kernel(const float* __restrict__ h2, const float* __restrict__ dinv,
                                 const float* __restrict__ b2, const float* __restrict__ agg2,
                                 float* __restrict__ out) {
    int t    = blockIdx.x * blockDim.x + threadIdx.x;
    int node = t >> 5;             // wave-uniform; 100000 % 8 == 0 -> full waves
    if (node >= N_NODES) return;
    int lane = t & 31;             // lane == class index (32 classes == wave32)
    float di  = dinv[node];
    int   idx = node * N_CLASSES + lane;
    float v   = agg2[idx] + h2[idx] * di * di + b2[lane];

    float m = v;
#pragma unroll
    for (int o = 16; o > 0; o >>= 1) m = fmaxf(m, __shfl_xor(m, o, 32));
    float s = expf(v - m);
#pragma unroll
    for (int o = 16; o > 0; o >>= 1) s += __shfl_xor(s, o, 32);
    out[idx] = (v - m) - logf(s);
}

// ---------------------------------------------------------------- launcher
extern "C" void kernel_launch(void* const* d_in, const int* in_sizes, int n_in,
                              void* d_out, int out_size, void* d_ws, size_t ws_size,
                              hipStream_t stream) {
    (void)in_sizes; (void)n_in; (void)out_size; (void)ws_size;
    const float* x  = (const float*)d_in[0];
    const int*   ei = (const int*)  d_in[1];
    const float* W1 = (const float*)d_in[2];
    const float* b1 = (const float*)d_in[3];
    const float* W2 = (const float*)d_in[4];
    const float* b2 = (const float*)d_in[5];
    float* out = (float*)d_out;

    // Workspace layout (floats): dinv[N] | region1[N*64] | region2[N*64]
    //   h1 = region1, agg1/out1 = region2
    //   h2 = region1[0 : N*32), agg2 = region1[N*32 : N*64)   (h1 dead by then)
    float* ws   = (float*)d_ws;
    float* dinv = ws;
    float* h1   = ws + N_NODES;
    float* agg1 = h1 + (size_t)N_NODES * HIDDEN;
    float* h2   = h1;
    float* agg2 = h1 + (size_t)N_NODES * N_CLASSES;

    const int B = 256;

    // degree (init 1.0 for self-loop) -> d^{-1/2}
    fill_kernel<<<(N_NODES + B - 1) / B, B, 0, stream>>>(dinv, 1.0f, N_NODES);
    degree_kernel<<<(N_EDGES + B - 1) / B, B, 0, stream>>>(ei + N_EDGES, dinv, N_EDGES);
    rsqrt_kernel<<<(N_NODES + B - 1) / B, B, 0, stream>>>(dinv, N_NODES);

    // ---- layer 1
    fill_kernel<<<(N_NODES * HIDDEN + B - 1) / B, B, 0, stream>>>(agg1, 0.0f, N_NODES * HIDDEN);
    gemm_wmma_kernel<F_IN, HIDDEN><<<(N_NODES + 31) / 32, B, 0, stream>>>(x, W1, h1, N_NODES);
    {
        long long th = (long long)N_EDGES * 32;
        scatter_kernel<HIDDEN><<<(int)((th + B - 1) / B), B, 0, stream>>>(ei, h1, dinv, agg1);
    }
    finalize1_kernel<<<(N_NODES * HIDDEN + B - 1) / B, B, 0, stream>>>(h1, dinv, b1, agg1);

    // ---- layer 2
    fill_kernel<<<(N_NODES * N_CLASSES + B - 1) / B, B, 0, stream>>>(agg2, 0.0f, N_NODES * N_CLASSES);
    gemm_wmma_kernel<HIDDEN, N_CLASSES><<<(N_NODES + 63) / 64, B, 0, stream>>>(agg1, W2, h2, N_NODES);
    {
        long long th = (long long)N_EDGES * 32;
        scatter_kernel<N_CLASSES><<<(int)((th + B - 1) / B), B, 0, stream>>>(ei, h2, dinv, agg2);
    }
    finalize2_logsoftmax_kernel<<<(N_NODES * 32 + B - 1) / B, B, 0, stream>>>(h2, dinv, b2, agg2, out);
}